// GIN_32890859552796
// MI455X (gfx1250) — compile-verified
//
#include <hip/hip_runtime.h>

// ---------------------------------------------------------------------------
// GIN forward for MI455X (gfx1250). FP32 throughout (memory-bound workload,
// ~1.4 GB HBM traffic dominates; GEMMs are 2.5 GFLOP and ride the WMMA pipe
// via V_WMMA_F32_16X16X4_F32 with exact f32 arithmetic).
//
// Round-2 change: weights staged into LDS in K-paired layout
//   sW[(k>>1)*128 + n*2 + (k&1)] = W[k][n]
// so each B fragment is one contiguous 8B-aligned v2f -> single ds_load_b64
// directly into the even-aligned VGPR pair WMMA consumes (removes the
// v_dual_mov shuffle traffic seen in round 1's assembly).
// ---------------------------------------------------------------------------

typedef float v2f __attribute__((ext_vector_type(2)));
typedef float v8f __attribute__((ext_vector_type(8)));

#define FEAT     64
#define HSTRIDE  68          // LDS row stride (floats) -> conflict-free A reads
#define BN_EPS   1e-5f

// --------------------------- small utility kernels -------------------------

__global__ void zero_f32v4(float4* __restrict__ p, int n4) {
    int i = blockIdx.x * blockDim.x + threadIdx.x;
    if (i < n4) p[i] = make_float4(0.f, 0.f, 0.f, 0.f);
}

// 16 threads per edge, one float4 each: coalesced 256B gather per edge row,
// scatter-add via global_atomic_add_f32.
__global__ void gin_scatter(const float* __restrict__ x,
                            const int* __restrict__ src,
                            const int* __restrict__ dst,
                            float* __restrict__ agg, int E) {
    int tid = blockIdx.x * blockDim.x + threadIdx.x;
    int e = tid >> 4;
    if (e >= E) return;
    int q = (tid & 15) << 2;
    int s = src[e];
    int d = dst[e];
    const float4 v = *reinterpret_cast<const float4*>(x + (size_t)s * FEAT + q);
    float* ap = agg + (size_t)d * FEAT + q;
    atomicAdd(ap + 0, v.x);
    atomicAdd(ap + 1, v.y);
    atomicAdd(ap + 2, v.z);
    atomicAdd(ap + 3, v.w);
}

// global_add_pool scatter: 16 threads per node.
__global__ void pool_scatter(const float* __restrict__ x,
                             const int* __restrict__ batch,
                             float* __restrict__ g, int n) {
    int tid = blockIdx.x * blockDim.x + threadIdx.x;
    int v = tid >> 4;
    if (v >= n) return;
    int q = (tid & 15) << 2;
    int b = batch[v];
    const float4 vv = *reinterpret_cast<const float4*>(x + (size_t)v * FEAT + q);
    float* gp = g + (size_t)b * FEAT + q;
    atomicAdd(gp + 0, vv.x);
    atomicAdd(gp + 1, vv.y);
    atomicAdd(gp + 2, vv.z);
    atomicAdd(gp + 3, vv.w);
}

// ----------------------- LDS staging in K-paired layout --------------------
// sW[(k>>1)*128 + n*2 + (k&1)] = W[k*64 + n]; 128 threads cooperate.
__device__ __forceinline__ void stage_weights_paired(float* sW,
                                                     const float* __restrict__ W,
                                                     int t) {
    for (int i = t; i < 64 * 64 / 4; i += 128) {
        int k  = i >> 4;            // source row (K index)
        int n4 = (i & 15) << 2;     // source col base
        float4 w = reinterpret_cast<const float4*>(W)[i];
        float* d = sW + (k >> 1) * 128 + (k & 1);
        d[(n4 + 0) * 2] = w.x;
        d[(n4 + 1) * 2] = w.y;
        d[(n4 + 2) * 2] = w.z;
        d[(n4 + 3) * 2] = w.w;
    }
}

// --------------------------- wave-level WMMA GEMM --------------------------
// One wave computes a 16x64 f32 tile:
//   D(16x64) = A(16x64, LDS, stride HSTRIDE) * B(64x64, LDS, K-paired layout)
// using 64x V_WMMA_F32_16X16X4_F32 (16 K-chunks x 4 N-tiles).
// A frag (16x4 f32): lanes 0-15 -> M=0..15 hold K=k0,k0+1 ; lanes 16-31 K=k0+2,k0+3
// B frag (4x16 f32): lanes 0-15 -> N=0..15 hold K=k0,k0+1 ; lanes 16-31 K=k0+2,k0+3
// C/D: VGPR i at lane L -> row = i + (L>=16 ? 8 : 0), col = (L & 15) + n0*16
__device__ __forceinline__ void wave_gemm64(const float* sA, const float* sB,
                                            v8f acc[4], int lane) {
    const int half2 = (lane >> 4) << 1;   // 0 or 2
    const int n16   = lane & 15;          // M for A-frag, N for B-frag
    for (int kc = 0; kc < 16; ++kc) {
        const int k = (kc << 2) + half2;          // even -> 8B aligned
        const v2f a = *reinterpret_cast<const v2f*>(&sA[n16 * HSTRIDE + k]);
        const float* brow = &sB[(k >> 1) * 128];  // paired K row
#pragma unroll
        for (int n0 = 0; n0 < 4; ++n0) {
            const v2f b = *reinterpret_cast<const v2f*>(
                &brow[((n0 << 4) + n16) * 2]);    // single ds_load_b64
            acc[n0] = __builtin_amdgcn_wmma_f32_16x16x4_f32(
                false, a, false, b, (short)0, acc[n0], false, false);
        }
    }
}

// --------------------------- fused GIN MLP kernel --------------------------
// Per block: 128 threads = 4 waves, 64 rows (16 per wave).
//   h  = (1+eps)*x + agg                       (staged in LDS)
//   y1 = relu( (h@W1 + b1 - mean)*rsqrt(var+e)*gamma + beta )   [folded]
//   y2 = y1@W2 + b2  -> xout
__global__ void __launch_bounds__(128)
gin_mlp(const float* __restrict__ xin, const float* __restrict__ agg,
        float* __restrict__ xout,
        const float* __restrict__ epsp,
        const float* __restrict__ W1, const float* __restrict__ b1,
        const float* __restrict__ gamma1, const float* __restrict__ beta1,
        const float* __restrict__ mean1, const float* __restrict__ var1,
        const float* __restrict__ W2, const float* __restrict__ b2,
        int n_rows) {
    __shared__ float sW1[64 * 64];
    __shared__ float sW2[64 * 64];
    __shared__ float sScale[64];
    __shared__ float sShift[64];
    __shared__ float sB2[64];
    __shared__ float sH[4][16 * HSTRIDE];

    const int t = threadIdx.x;

    stage_weights_paired(sW1, W1, t);
    stage_weights_paired(sW2, W2, t);
    if (t < 64) {
        float sc = gamma1[t] * rsqrtf(var1[t] + BN_EPS);
        sScale[t] = sc;
        sShift[t] = beta1[t] + (b1[t] - mean1[t]) * sc;  // folds b1 + BN shift
        sB2[t]   = b2[t];
    }
    const float onePlusEps = 1.0f + epsp[0];

    const int wave = t >> 5;
    const int lane = t & 31;
    const int r0   = blockIdx.x * 64 + wave * 16;
    float* hw = sH[wave];

    // build h tile: 16 rows x 64 cols (8 float4 slots per lane)
#pragma unroll
    for (int i = 0; i < 8; ++i) {
        int s  = i * 32 + lane;
        int m  = s >> 4;
        int c4 = (s & 15) << 2;
        int r  = r0 + m;
        float4 v = make_float4(0.f, 0.f, 0.f, 0.f);
        if (r < n_rows) {
            float4 xv = *reinterpret_cast<const float4*>(xin + (size_t)r * FEAT + c4);
            float4 av = *reinterpret_cast<const float4*>(agg + (size_t)r * FEAT + c4);
            v.x = onePlusEps * xv.x + av.x;
            v.y = onePlusEps * xv.y + av.y;
            v.z = onePlusEps * xv.z + av.z;
            v.w = onePlusEps * xv.w + av.w;
        }
        float* hp = hw + m * HSTRIDE + c4;
        hp[0] = v.x; hp[1] = v.y; hp[2] = v.z; hp[3] = v.w;
    }
    __syncthreads();

    // GEMM1: h @ W1
    v8f acc[4] = {};
    wave_gemm64(hw, sW1, acc, lane);

    // bias+BN+ReLU, overwrite wave-private LDS tile (per-wave, LDS in-order)
    const int n16  = lane & 15;
    const int half = lane >> 4;
#pragma unroll
    for (int n0 = 0; n0 < 4; ++n0) {
        int col = (n0 << 4) + n16;
        float sc = sScale[col];
        float sh = sShift[col];
#pragma unroll
        for (int i = 0; i < 8; ++i) {
            int row = i + (half << 3);
            float y = acc[n0][i] * sc + sh;
            hw[row * HSTRIDE + col] = fmaxf(y, 0.f);
        }
    }

    // GEMM2: y1 @ W2, add b2, write out
    v8f acc2[4] = {};
    wave_gemm64(hw, sW2, acc2, lane);
#pragma unroll
    for (int n0 = 0; n0 < 4; ++n0) {
        int col = (n0 << 4) + n16;
        float bb = sB2[col];
#pragma unroll
        for (int i = 0; i < 8; ++i) {
            int row = i + (half << 3);
            int r = r0 + row;
            if (r < n_rows) xout[(size_t)r * FEAT + col] = acc2[n0][i] + bb;
        }
    }
}

// --------------------------- pooled linear (WMMA) --------------------------
// out = relu(A @ W + bias), A is [n_rows x 64], W [64x64].
__global__ void __launch_bounds__(128)
gemm_bias_relu(const float* __restrict__ A, const float* __restrict__ W,
               const float* __restrict__ bias, float* __restrict__ out,
               int n_rows) {
    __shared__ float sW[64 * 64];
    __shared__ float sB[64];
    __shared__ float sH[4][16 * HSTRIDE];

    const int t = threadIdx.x;
    stage_weights_paired(sW, W, t);
    if (t < 64) sB[t] = bias[t];

    const int wave = t >> 5;
    const int lane = t & 31;
    const int r0   = blockIdx.x * 64 + wave * 16;
    float* hw = sH[wave];

#pragma unroll
    for (int i = 0; i < 8; ++i) {
        int s  = i * 32 + lane;
        int m  = s >> 4;
        int c4 = (s & 15) << 2;
        int r  = r0 + m;
        float4 v = make_float4(0.f, 0.f, 0.f, 0.f);
        if (r < n_rows)
            v = *reinterpret_cast<const float4*>(A + (size_t)r * FEAT + c4);
        float* hp = hw + m * HSTRIDE + c4;
        hp[0] = v.x; hp[1] = v.y; hp[2] = v.z; hp[3] = v.w;
    }
    __syncthreads();

    v8f acc[4] = {};
    wave_gemm64(hw, sW, acc, lane);

    const int n16  = lane & 15;
    const int half = lane >> 4;
#pragma unroll
    for (int n0 = 0; n0 < 4; ++n0) {
        int col = (n0 << 4) + n16;
        float bb = sB[col];
#pragma unroll
        for (int i = 0; i < 8; ++i) {
            int row = i + (half << 3);
            int r = r0 + row;
            if (r < n_rows)
                out[(size_t)r * FEAT + col] = fmaxf(acc[n0][i] + bb, 0.f);
        }
    }
}

// final 64 -> 2 projection: one thread per output element.
__global__ void final_proj(const float* __restrict__ gh,
                           const float* __restrict__ Wf,
                           const float* __restrict__ bf,
                           float* __restrict__ out, int n_out) {
    int i = blockIdx.x * blockDim.x + threadIdx.x;
    if (i >= n_out) return;
    int gi = i >> 1;
    int c  = i & 1;
    float s = bf[c];
    const float* row = gh + (size_t)gi * FEAT;
#pragma unroll
    for (int k = 0; k < 64; ++k) s += row[k] * Wf[k * 2 + c];
    out[i] = s;
}

// ------------------------------- launcher ----------------------------------

extern "C" void kernel_launch(void* const* d_in, const int* in_sizes, int n_in,
                              void* d_out, int out_size, void* d_ws, size_t ws_size,
                              hipStream_t stream) {
    const float* x     = (const float*)d_in[0];
    const int*   ei    = (const int*)d_in[1];
    const int*   batch = (const int*)d_in[2];

    const int N = in_sizes[2];          // 50000 nodes
    const int E = in_sizes[1] / 2;      // 800000 edges
    const int G = out_size / 2;         // 512 graphs
    const int* src = ei;
    const int* dst = ei + E;

    // params flattened in dict-insertion order:
    //   convs[l]: eps,W1,b1,gamma1,beta1,mean1,var1,W2,b2  (9 entries each, l=0..2)
    //   then lin_W, lin_b, final_W, final_b
    auto inF = [&](int i) { return (const float*)d_in[i]; };

    float* wsX   = (float*)d_ws;                       // N*64
    float* wsAgg = wsX + (size_t)N * FEAT;             // N*64
    float* wsG   = wsAgg + (size_t)N * FEAT;           // G*64
    float* wsGh  = wsG + (size_t)G * FEAT;             // G*64

    const float* cur = x;
    for (int l = 0; l < 3; ++l) {
        const int base = 3 + l * 9;
        // zero aggregation buffer
        {
            int n4 = N * FEAT / 4;
            zero_f32v4<<<(n4 + 255) / 256, 256, 0, stream>>>((float4*)wsAgg, n4);
        }
        // edge scatter-add
        {
            long long th = (long long)E * 16;
            gin_scatter<<<(unsigned)((th + 255) / 256), 256, 0, stream>>>(
                cur, src, dst, wsAgg, E);
        }
        // fused (1+eps)x+agg -> Linear -> BN -> ReLU -> Linear
        gin_mlp<<<(N + 63) / 64, 128, 0, stream>>>(
            cur, wsAgg, wsX,
            inF(base + 0), inF(base + 1), inF(base + 2), inF(base + 3),
            inF(base + 4), inF(base + 5), inF(base + 6), inF(base + 7),
            inF(base + 8), N);
        cur = wsX;
    }

    // global_add_pool
    {
        int n4 = G * FEAT / 4;
        zero_f32v4<<<(n4 + 255) / 256, 256, 0, stream>>>((float4*)wsG, n4);
        long long th = (long long)N * 16;
        pool_scatter<<<(unsigned)((th + 255) / 256), 256, 0, stream>>>(
            cur, batch, wsG, N);
    }

    // relu(g @ lin_W + lin_b)
    gemm_bias_relu<<<(G + 63) / 64, 128, 0, stream>>>(
        wsG, inF(30), inF(31), wsGh, G);

    // @ final_W + final_b -> [G,2]
    final_proj<<<(out_size + 255) / 256, 256, 0, stream>>>(
        wsGh, inF(32), inF(33), (float*)d_out, out_size);
}